// CnUpdateLayer_19825569038600
// MI455X (gfx1250) — compile-verified
//
#include <hip/hip_runtime.h>
#include <math.h>

typedef __attribute__((ext_vector_type(2))) float v2f;
typedef __attribute__((ext_vector_type(8))) float v8f;

#define B_DIM  32
#define IN_F   2048
#define OUT_F  2048
#define KSPLIT 4
#define KCHUNK (IN_F / KSPLIT)   // 512 K-elements per split

// Pass 1: transform x[32][2048] into two A-planes in workspace:
//   p1[b][j] = log|x|  (0 when x==0, so the term vanishes in the sum)
//   p2[b][j] = (x<0 ? 1 : 0) + (x!=0 ? 4096 : 0)   (parity + nonzero count, packed)
__global__ void cn_prep_kernel(const float* __restrict__ x,
                               float* __restrict__ p1,
                               float* __restrict__ p2) {
    int i = blockIdx.x * blockDim.x + threadIdx.x;
    if (i < B_DIM * IN_F) {
        float v  = x[i];
        bool  nz = (v != 0.0f);
        p1[i] = nz ? logf(fabsf(v)) : 0.0f;
        p2[i] = (v < 0.0f ? 1.0f : 0.0f) + (nz ? 4096.0f : 0.0f);
    }
}

// Pass 2: WMMA f32 16x16x4 partial matmuls, two channels sharing the mask (B)
// operand. One wave computes a 32x16 output tile (2 M-tiles of 16x16) over one
// K-chunk of 512. blockIdx.x = N tile group, blockIdx.y = K split.
__global__ __launch_bounds__(128) void cn_wmma_partial(const float* __restrict__ p1,
                                                       const float* __restrict__ p2,
                                                       const float* __restrict__ mask,
                                                       float* __restrict__ part) {
    const int lane  = threadIdx.x & 31;
    const int wave  = threadIdx.x >> 5;
    const int n0    = (blockIdx.x * 4 + wave) * 16;   // 16 output columns per wave
    const int k0    = blockIdx.y * KCHUNK;            // K range of this split
    const int l     = lane & 15;
    const int koff  = (lane < 16) ? 0 : 2;            // half-wave K split of the 16x4 fragment

    // B fragment source: B[k, n] = mask[n*IN_F + k] -> contiguous along K per lane.
    const float* bp   = mask + (size_t)(n0 + l) * IN_F + k0 + koff;
    // A fragment sources for the two channels, two 16-row M tiles each.
    const float* a1t0 = p1 + (size_t)l        * IN_F + k0 + koff;
    const float* a1t1 = p1 + (size_t)(l + 16) * IN_F + k0 + koff;
    const float* a2t0 = p2 + (size_t)l        * IN_F + k0 + koff;
    const float* a2t1 = p2 + (size_t)(l + 16) * IN_F + k0 + koff;

    v8f c1t0 = {}; v8f c1t1 = {};   // log-magnitude accumulators
    v8f c2t0 = {}; v8f c2t1 = {};   // packed parity/nonzero accumulators

    for (int k = 0; k < KCHUNK; k += 4) {
        v2f bf  = *(const v2f*)(bp   + k);   // shared mask fragment
        v2f f10 = *(const v2f*)(a1t0 + k);
        v2f f11 = *(const v2f*)(a1t1 + k);
        v2f f20 = *(const v2f*)(a2t0 + k);
        v2f f21 = *(const v2f*)(a2t1 + k);
        c1t0 = __builtin_amdgcn_wmma_f32_16x16x4_f32(false, f10, false, bf, (short)0, c1t0, false, false);
        c1t1 = __builtin_amdgcn_wmma_f32_16x16x4_f32(false, f11, false, bf, (short)0, c1t1, false, false);
        c2t0 = __builtin_amdgcn_wmma_f32_16x16x4_f32(false, f20, false, bf, (short)0, c2t0, false, false);
        c2t1 = __builtin_amdgcn_wmma_f32_16x16x4_f32(false, f21, false, bf, (short)0, c2t1, false, false);
    }

    // Store partials. C/D layout: VGPR v holds (M=v, N=lane) for lanes 0-15 and
    // (M=v+8, N=lane-16) for lanes 16-31.
    // part layout: [split][channel][B_DIM][OUT_F]
    const int    nc    = n0 + l;
    const int    mbase = (lane < 16) ? 0 : 8;
    const size_t plane = (size_t)B_DIM * OUT_F;
    float* pc1 = part + (size_t)(blockIdx.y * 2 + 0) * plane;
    float* pc2 = part + (size_t)(blockIdx.y * 2 + 1) * plane;
#pragma unroll
    for (int v = 0; v < 8; ++v) {
        pc1[(size_t)(mbase + v)      * OUT_F + nc] = c1t0[v];
        pc1[(size_t)(16 + mbase + v) * OUT_F + nc] = c1t1[v];
        pc2[(size_t)(mbase + v)      * OUT_F + nc] = c2t0[v];
        pc2[(size_t)(16 + mbase + v) * OUT_F + nc] = c2t1[v];
    }
}

// Pass 3: combine K-split partials and apply sign/parity/exp epilogue.
__global__ void cn_combine_kernel(const float* __restrict__ part,
                                  float* __restrict__ out) {
    int i = blockIdx.x * blockDim.x + threadIdx.x;
    if (i < B_DIM * OUT_F) {
        const size_t plane = (size_t)B_DIM * OUT_F;
        float s1 = 0.0f, s2 = 0.0f;
#pragma unroll
        for (int s = 0; s < KSPLIT; ++s) {
            s1 += part[(size_t)(s * 2 + 0) * plane + i];
            s2 += part[(size_t)(s * 2 + 1) * plane + i];   // exact integer sums
        }
        int   si  = (int)(s2 + 0.5f);
        float sg  = (si & 1) ? -1.0f : 1.0f;               // parity of negatives
        out[i] = (si >= 4096) ? sg * expf(s1) : 0.0f;      // any-nonzero gate
    }
}

extern "C" void kernel_launch(void* const* d_in, const int* in_sizes, int n_in,
                              void* d_out, int out_size, void* d_ws, size_t ws_size,
                              hipStream_t stream) {
    const float* x    = (const float*)d_in[0];   // [32, 2048] f32
    const float* mask = (const float*)d_in[1];   // [2048, 2048] f32 (0/1)

    float* p1   = (float*)d_ws;                  // [32, 2048]
    float* p2   = p1 + B_DIM * IN_F;             // [32, 2048]
    float* part = p2 + B_DIM * IN_F;             // [KSPLIT][2][32][2048] = 2 MB

    cn_prep_kernel<<<(B_DIM * IN_F + 255) / 256, 256, 0, stream>>>(x, p1, p2);

    dim3 grid(OUT_F / (4 * 16), KSPLIT);         // 32 N-groups x 4 K-splits = 512 waves
    cn_wmma_partial<<<grid, 128, 0, stream>>>(p1, p2, mask, part);

    cn_combine_kernel<<<(B_DIM * OUT_F + 255) / 256, 256, 0, stream>>>(part, (float*)d_out);
}